// SAGE_50362786513304
// MI455X (gfx1250) — compile-verified
//
#include <hip/hip_runtime.h>

typedef __attribute__((ext_vector_type(2))) float v2f;
typedef __attribute__((ext_vector_type(8))) float v8f;

#define S0n 1024000
#define D0n 102400
#define D1n 10240
#define D2n 1024
#define E0n 1024000
#define E1n 102400
#define E2n 10240
#define INn 128
#define HIDn 256
#define OUTn 47

static __device__ __forceinline__ v8f wmma4(v2f a, v2f b, v8f c) {
    // D = A(16x4 f32) * B(4x16 f32) + C(16x16 f32), exact fp32 WMMA
    return __builtin_amdgcn_wmma_f32_16x16x4_f32(false, a, false, b, (short)0, c,
                                                 false, false);
}

// --- fused scatter: ssum[dst[e], f..f+3] += H[src[e], f..f+3]; also degree count ---
// One thread handles 4 contiguous features (global_load_b128 + 4 f32 atomics).
__global__ void scatter_kernel(const float* __restrict__ H, const int* __restrict__ src,
                               const int* __restrict__ dst, float* __restrict__ ssum,
                               float* __restrict__ cnt, long long total, int fshift) {
    long long idx = (long long)blockIdx.x * blockDim.x + threadIdx.x;
    if (idx >= total) return;                 // total = E * (F/4)
    int qmask = (1 << (fshift - 2)) - 1;
    long long e = idx >> (fshift - 2);
    int q = (int)(idx & qmask);               // which float4 within the row
    int f = q << 2;
    int s = src[e];
    int d = dst[e];
    const float4 v = *(const float4*)(H + (((long long)s) << fshift) + f);
    float* p = ssum + (((long long)d) << fshift) + f;
    atomicAdd(p + 0, v.x);
    atomicAdd(p + 1, v.y);
    atomicAdd(p + 2, v.z);
    atomicAdd(p + 3, v.w);
    if (q == 0) atomicAdd(cnt + d, 1.0f);
}

// --- fused SAGE GEMM: Out = act(Aself@Wself + (Ssum/max(cnt,1))@Wneigh + b) ---
// Block = 8 waves. blockIdx.y = one 16-col N tile; blockIdx.x*8+wave = 16-row M tile.
// One wave = one 16x16 tile with a single v8f accumulator: low register pressure lets
// the compiler keep ~16 global_load_b64 in flight (staggered s_wait_loadcnt).
// Weight slices are transposed into LDS as [matrix][col][k] with stride K+4 floats:
//   bank(col,k) = (4*col + klo + k) mod 64  -> all 64 banks distinct for ds_load_b64.
// N-tail (N=47) masked once at LDS-fill time (zero fill), so hot loop has no masks.
// WMMA VGPR layouts per ISA: A lane l -> row (l&15), K pair (l>>4)*2 (contiguous v2f);
// B lane l -> col (l&15), same K pair; C VGPR r -> row r + 8*(l>>4), col (l&15).
template <int K, bool RELU>
__global__ __launch_bounds__(256) void sage_gemm_kernel(
    const float* __restrict__ Aself, const float* __restrict__ Ssum,
    const float* __restrict__ Cnt, const float* __restrict__ Wself,
    const float* __restrict__ Wneigh, const float* __restrict__ Bias,
    float* __restrict__ Out, int M, int N) {
    constexpr int LSTR = K + 4;  // LDS column stride in floats (bank-conflict-free)
    __shared__ float lds[2 * 16 * LSTR];

    const int tid = threadIdx.x;
    const int colBase = blockIdx.y << 4;

    // Cooperative transpose of both weight slices into LDS (masked zero-fill past N).
    for (int i = tid; i < 2 * 16 * K; i += 256) {
        int m = i >> (4 + __builtin_ctz(K));        // 0 = self, 1 = neigh
        int r = i & (16 * K - 1);
        int c = r >> __builtin_ctz(K);              // 0..15
        int k = r & (K - 1);
        int gc = colBase + c;
        const float* W = m ? Wneigh : Wself;
        lds[(m * 16 + c) * LSTR + k] = (gc < N) ? W[k * N + gc] : 0.0f;
    }
    __syncthreads();

    const int wave = tid >> 5;
    const int lane = tid & 31;
    const int tm = (blockIdx.x << 3) + wave;
    if (tm >= (M >> 4)) return;  // wave-uniform: EXEC all-ones at every WMMA

    const int mrow = (tm << 4) + (lane & 15);
    const int col = colBase + (lane & 15);
    const int klo = (lane >> 4) << 1;  // 0 or 2

    const float inv = 1.0f / fmaxf(Cnt[mrow], 1.0f);

    const float* arow = Aself + (size_t)mrow * K + klo;
    const float* nrow = Ssum + (size_t)mrow * K + klo;
    const float* bs = lds + (lane & 15) * LSTR + klo;
    const float* bn = bs + 16 * LSTR;

    v8f acc = {};
#pragma unroll 8
    for (int k = 0; k < K; k += 4) {
        v2f a = *(const v2f*)(arow + k);
        v2f b = *(const v2f*)(bs + k);
        acc = wmma4(a, b, acc);
        v2f an = *(const v2f*)(nrow + k);
        an.x *= inv;
        an.y *= inv;
        v2f b2 = *(const v2f*)(bn + k);
        acc = wmma4(an, b2, acc);
    }

    if (col < N) {
        float bv = Bias[col];
        int rbase = (tm << 4) + ((lane >> 4) << 3);
#pragma unroll
        for (int r = 0; r < 8; ++r) {
            float v = acc[r] + bv;
            if (RELU) v = fmaxf(v, 0.0f);
            Out[(long long)(rbase + r) * N + col] = v;
        }
    }
}

extern "C" void kernel_launch(void* const* d_in, const int* in_sizes, int n_in,
                              void* d_out, int out_size, void* d_ws, size_t ws_size,
                              hipStream_t stream) {
    const float* h       = (const float*)d_in[0];
    const int*   src0    = (const int*)d_in[1];
    const int*   dst0    = (const int*)d_in[2];
    const int*   src1    = (const int*)d_in[3];
    const int*   dst1    = (const int*)d_in[4];
    const int*   src2    = (const int*)d_in[5];
    const int*   dst2    = (const int*)d_in[6];
    const float* Wself0  = (const float*)d_in[7];
    const float* Wneigh0 = (const float*)d_in[8];
    const float* b0      = (const float*)d_in[9];
    const float* Wself1  = (const float*)d_in[10];
    const float* Wneigh1 = (const float*)d_in[11];
    const float* b1      = (const float*)d_in[12];
    const float* Wself2  = (const float*)d_in[13];
    const float* Wneigh2 = (const float*)d_in[14];
    const float* b2      = (const float*)d_in[15];

    // Workspace layout (floats). Accumulators first so one memset clears them.
    float* ws    = (float*)d_ws;
    float* ssum0 = ws;                          // D0*IN
    float* cnt0  = ssum0 + (size_t)D0n * INn;   // D0
    float* ssum1 = cnt0 + D0n;                  // D1*HID
    float* cnt1  = ssum1 + (size_t)D1n * HIDn;  // D1
    float* ssum2 = cnt1 + D1n;                  // D2*HID
    float* cnt2  = ssum2 + (size_t)D2n * HIDn;  // D2
    float* aend  = cnt2 + D2n;
    float* h1    = aend;                        // D0*HID
    float* h2    = h1 + (size_t)D0n * HIDn;     // D1*HID

    size_t accumBytes = (size_t)(aend - ws) * sizeof(float);
    hipMemsetAsync(d_ws, 0, accumBytes, stream);

    // ---- layer 0: scatter-mean over E0 edges (F=128), fused WMMA GEMM + ReLU ----
    {
        long long total = (long long)E0n * (INn / 4);
        scatter_kernel<<<(int)((total + 255) / 256), 256, 0, stream>>>(
            h, src0, dst0, ssum0, cnt0, total, 7);
    }
    {
        dim3 grid((D0n / 16 + 7) / 8, HIDn / 16);
        sage_gemm_kernel<INn, true><<<grid, 256, 0, stream>>>(
            h, ssum0, cnt0, Wself0, Wneigh0, b0, h1, D0n, HIDn);
    }

    // ---- layer 1: F=256 ----
    {
        long long total = (long long)E1n * (HIDn / 4);
        scatter_kernel<<<(int)((total + 255) / 256), 256, 0, stream>>>(
            h1, src1, dst1, ssum1, cnt1, total, 8);
    }
    {
        dim3 grid((D1n / 16 + 7) / 8, HIDn / 16);
        sage_gemm_kernel<HIDn, true><<<grid, 256, 0, stream>>>(
            h1, ssum1, cnt1, Wself1, Wneigh1, b1, h2, D1n, HIDn);
    }

    // ---- layer 2: F=256, N=47 (masked at LDS fill), no ReLU, write d_out ----
    {
        long long total = (long long)E2n * (HIDn / 4);
        scatter_kernel<<<(int)((total + 255) / 256), 256, 0, stream>>>(
            h2, src2, dst2, ssum2, cnt2, total, 8);
    }
    {
        dim3 grid((D2n / 16 + 7) / 8, (OUTn + 15) / 16);
        sage_gemm_kernel<HIDn, false><<<grid, 256, 0, stream>>>(
            h2, ssum2, cnt2, Wself2, Wneigh2, b2, (float*)d_out, D2n, OUTn);
    }
}